// Wasserstein_Sinkhorn_69320772157808
// MI455X (gfx1250) — compile-verified
//
#include <hip/hip_runtime.h>
#include <math.h>

typedef __attribute__((ext_vector_type(2))) float v2f;
typedef __attribute__((ext_vector_type(4))) float v4f;
typedef __attribute__((ext_vector_type(8))) float v8f;

#define NPT     256            // points per cloud (= NT, SPARSE=1)
#define NRCV    64
#define NSEG    4
#define NBATCH  2
#define NSTEPS  24
#define DT_S    0.001f
#define EPSMIN  1e-4f          // BLUR**P
#define RATIO   0.25f          // SCALING**P
#define LOGN    5.545177444479562f   // ln(256)
#define LOG2E   1.4426950408889634f
#define LN2     0.6931471805599453f

__device__ __forceinline__ float wave_sum(float v) {
#pragma unroll
  for (int d = 1; d < 32; d <<= 1) v += __shfl_xor(v, d, 32);
  return v;
}

__global__ __launch_bounds__(256)
void sinkhorn_kernel(const float* __restrict__ syn, const float* __restrict__ obs,
                     float* __restrict__ ws) {
  // per-problem LDS working set (~29 KB)
  __shared__ float aO[NPT], aS[NPT];      // obs / syn amplitudes
  __shared__ float sqO[NPT], sqS[NPT];    // |x_i|^2 per point (rows side)
  __shared__ float pot[2][4][NPT];        // double-buffered f,g,px,py
  __shared__ v4f   ccol[4][NPT];          // per-step packed columns {t, amp, ct, 0}
  __shared__ float red[64];

  const int p    = blockIdx.x;            // problem id in [0, B*S*NR)
  const int bs   = p / NRCV;              // b*S + s
  const int r    = p % NRCV;
  const int tid  = threadIdx.x;
  const int lane = tid & 31;
  const int wv   = tid >> 5;

  // ---- load one (b,s,:,r) trace pair; strided gather, tiny vs compute ----
  const float o  = obs[((size_t)bs * NPT + tid) * NRCV + r];
  const float sy = syn[((size_t)bs * NPT + tid) * NRCV + r];
  const float t  = (float)tid * DT_S;
  const float so = t * t + o * o;
  const float ss = t * t + sy * sy;
  aO[tid] = o;  aS[tid] = sy;
  sqO[tid] = so; sqS[tid] = ss;
  pot[0][0][tid] = 0.f; pot[0][1][tid] = 0.f;
  pot[0][2][tid] = 0.f; pot[0][3][tid] = 0.f;

  // ---- diameter + all-zero-trace mask reductions ----
  float mx = fmaxf(o, sy), mn = fminf(o, sy);
  float sao = fabsf(o), sas = fabsf(sy);
#pragma unroll
  for (int d = 1; d < 32; d <<= 1) {
    mx  = fmaxf(mx, __shfl_xor(mx, d, 32));
    mn  = fminf(mn, __shfl_xor(mn, d, 32));
    sao += __shfl_xor(sao, d, 32);
    sas += __shfl_xor(sas, d, 32);
  }
  if (lane == 0) { red[wv] = mx; red[8 + wv] = mn; red[16 + wv] = sao; red[24 + wv] = sas; }
  __syncthreads();
  if (tid == 0) {
    float MX = red[0], MN = red[8], SO = red[16], SS = red[24];
    for (int i = 1; i < 8; ++i) {
      MX = fmaxf(MX, red[i]); MN = fminf(MN, red[8 + i]);
      SO += red[16 + i];      SS += red[24 + i];
    }
    const float diam = fmaxf((float)(NPT - 1) * DT_S, MX - MN);
    red[32] = diam * diam;                          // eps0 = diameter^P
    red[33] = (SO == 0.f && SS == 0.f) ? 0.f : 1.f; // mask keep flag
  }
  __syncthreads();
  const float eps0 = red[32];
  const float keep = red[33];

  const bool lo    = lane < 16;
  const int  l15   = lane & 15;
  const int  half8 = (lane >= 16) ? 8 : 0;

  float e = eps0;
  for (int s = 0; s < NSTEPS; ++s) {
    const float eps = fmaxf(e, EPSMIN);
    e *= RATIO;
    const float k  = LOG2E / eps;   // exp(w/eps) = exp2(w*k)
    const int po = s & 1, pn = po ^ 1;

    // ---- per-step packed column tables: {t_n, amp_n, pot_n - 0.5*|y_n|^2, 0} ----
    {
      const float f_  = pot[po][0][tid];
      const float g_  = pot[po][1][tid];
      const float px_ = pot[po][2][tid];
      const float py_ = pot[po][3][tid];
      ccol[0][tid] = (v4f){t, sy, g_  - 0.5f * ss, 0.f};  // f  <- Cxy: cols=syn, pot=g
      ccol[1][tid] = (v4f){t, o,  f_  - 0.5f * so, 0.f};  // g  <- Cyx: cols=obs, pot=f
      ccol[2][tid] = (v4f){t, o,  px_ - 0.5f * so, 0.f};  // px <- Cxx
      ccol[3][tid] = (v4f){t, sy, py_ - 0.5f * ss, 0.f};  // py <- Cyy
    }
    __syncthreads();

    // 64 tile-row tasks per step: task = matrix*16 + rowblock
#pragma unroll 1
    for (int tt = 0; tt < 8; ++tt) {
      const int task = tt * 8 + wv;     // wave-uniform => EXEC all-1s at WMMA
      const int m  = task >> 4;
      const int rb = task & 15;

      const float *rowA, *rowSq, *pin;
      int outIdx; bool damp;
      switch (m) {
        case 0:  rowA=aO; rowSq=sqO; pin=pot[po][0]; outIdx=0; damp=false; break; // rows=obs
        case 1:  rowA=aS; rowSq=sqS; pin=pot[po][1]; outIdx=1; damp=false; break; // rows=syn
        case 2:  rowA=aO; rowSq=sqO; pin=pot[po][2]; outIdx=2; damp=true;  break; // rows=obs
        default: rowA=aS; rowSq=sqS; pin=pot[po][3]; outIdx=3; damp=true;  break; // rows=syn
      }
      const v4f* cc = ccol[m];

      const int rowbase = rb * 16;
      // A fragment: 16x4 f32. K=2,3 (lanes 16-31) are ZERO, which makes the
      // corresponding B rows don't-care -> no selects needed on the B side.
      const float ra = rowA[rowbase + l15];
      v2f afrag;
      afrag.x = lo ? (float)(rowbase + l15) * DT_S : 0.f;
      afrag.y = lo ? ra                            : 0.f;

      float rt[8], mrow[8], srow[8];
#pragma unroll
      for (int rr = 0; rr < 8; ++rr) {
        rt[rr]   = 0.5f * rowSq[rowbase + half8 + rr];
        mrow[rr] = -1e30f;
      }

      // ---------- pass 1: exact row maxima of w = pot[N] - C[M][N] ----------
      for (int jb = 0; jb < 16; ++jb) {
        const v4f c = cc[jb * 16 + l15];     // one ds_load_b128 per tile
        v2f bfrag; bfrag.x = c.x; bfrag.y = c.y;
        const float ct = c.z;

        v8f acc = {0.f, 0.f, 0.f, 0.f, 0.f, 0.f, 0.f, 0.f};
        acc = __builtin_amdgcn_wmma_f32_16x16x4_f32(
            false, afrag, false, bfrag, (short)0, acc, false, false);

#pragma unroll
        for (int rr = 0; rr < 8; ++rr)
          mrow[rr] = fmaxf(mrow[rr], acc[rr] + (ct - rt[rr]));
      }
      // merge maxima across the 16 lanes holding each row
#pragma unroll
      for (int rr = 0; rr < 8; ++rr) {
#pragma unroll
        for (int d = 1; d < 16; d <<= 1)
          mrow[rr] = fmaxf(mrow[rr], __shfl_xor(mrow[rr], d, 32));
      }

      // per-row folded constant for pass 2: arg = acc*k + (ct*k - rk[rr])
      float rk[8];
#pragma unroll
      for (int rr = 0; rr < 8; ++rr) { rk[rr] = (rt[rr] + mrow[rr]) * k; srow[rr] = 0.f; }

      // ---------- pass 2: sums of exp2((w - M)*k), one exp per element ----------
      for (int jb = 0; jb < 16; ++jb) {
        const v4f c = cc[jb * 16 + l15];
        v2f bfrag; bfrag.x = c.x; bfrag.y = c.y;
        const float ctk = c.z * k;

        v8f acc = {0.f, 0.f, 0.f, 0.f, 0.f, 0.f, 0.f, 0.f};
        acc = __builtin_amdgcn_wmma_f32_16x16x4_f32(
            false, afrag, false, bfrag, (short)0, acc, false, false);

#pragma unroll
        for (int rr = 0; rr < 8; ++rr)
          srow[rr] += __builtin_amdgcn_exp2f(fmaf(acc[rr], k, ctk - rk[rr]));
      }
      // sum partial row sums across the 16 lanes of each half
#pragma unroll
      for (int rr = 0; rr < 8; ++rr) {
#pragma unroll
        for (int d = 1; d < 16; d <<= 1)
          srow[rr] += __shfl_xor(srow[rr], d, 32);
      }

      if (l15 == 0) {   // lanes 0 and 16 own rows M..M+7 / M+8..M+15
#pragma unroll
        for (int rr = 0; rr < 8; ++rr) {
          const int M = rowbase + half8 + rr;
          // softmin = -eps*la - max - eps*ln(sum) ; la = -ln(n)
          float res = eps * LOGN - mrow[rr]
                    - eps * LN2 * __builtin_amdgcn_logf(srow[rr]);
          if (damp) res = 0.5f * (pin[M] + res);   // symmetric damped update
          pot[pn][outIdx][M] = res;
        }
      }
    }
    __syncthreads();
  }

  // NSTEPS even -> final potentials in parity 0
  float val = (pot[0][0][tid] - pot[0][2][tid] +
               pot[0][1][tid] - pot[0][3][tid]) * (1.0f / (float)NPT);
  val = wave_sum(val);
  if (lane == 0) red[wv] = val;
  __syncthreads();
  if (tid == 0) {
    float ttl = 0.f;
    for (int i = 0; i < 8; ++i) ttl += red[i];
    ws[p] = keep * ttl;
  }
}

// deterministic per-batch sum of the 256 per-problem divergences
__global__ __launch_bounds__(256)
void reduce_kernel(const float* __restrict__ ws, float* __restrict__ out) {
  __shared__ float red[8];
  const int b = blockIdx.x;
  float v = ws[b * (NSEG * NRCV) + threadIdx.x];
  v = wave_sum(v);
  if ((threadIdx.x & 31) == 0) red[threadIdx.x >> 5] = v;
  __syncthreads();
  if (threadIdx.x == 0) {
    float t = 0.f;
    for (int i = 0; i < 8; ++i) t += red[i];
    out[b] = t;
  }
}

extern "C" void kernel_launch(void* const* d_in, const int* in_sizes, int n_in,
                              void* d_out, int out_size, void* d_ws, size_t ws_size,
                              hipStream_t stream) {
  (void)in_sizes; (void)n_in; (void)ws_size;
  const float* syn = (const float*)d_in[0];
  const float* obs = (const float*)d_in[1];
  float* out = (float*)d_out;
  float* ws  = (float*)d_ws;   // 512 floats of scratch

  sinkhorn_kernel<<<NBATCH * NSEG * NRCV, 256, 0, stream>>>(syn, obs, ws);
  reduce_kernel<<<out_size, 256, 0, stream>>>(ws, out);
}